// dc_loss_74019466379798
// MI455X (gfx1250) — compile-verified
//
#include <hip/hip_runtime.h>
#include <hip/hip_bf16.h>

#define HH 512
#define WW 512
#define HWsz (HH * WW)
#define TILE 64
#define PAD 17
#define EXT 98              // TILE + 2*PAD
#define HBW 94              // EXT - 4 (5-wide block positions)
#define NPART 1024          // 16 pairs * 64 tiles

// ---- CDNA5 async global->LDS path (guarded so compile stays green) ----
#if defined(__HIP_DEVICE_COMPILE__) && __has_builtin(__builtin_amdgcn_global_load_async_to_lds_b32)
#define HAVE_ASYNC 1
#else
#define HAVE_ASYNC 0
#endif

#if HAVE_ASYNC
typedef __attribute__((address_space(1))) int GInt;
typedef __attribute__((address_space(3))) int LInt;
__device__ __forceinline__ void async_cp_b32(const void* g, void* l) {
  __builtin_amdgcn_global_load_async_to_lds_b32((GInt*)g, (LInt*)l, 0, 0);
}
__device__ __forceinline__ void wait_async0() {
#if __has_builtin(__builtin_amdgcn_s_wait_asynccnt)
  __builtin_amdgcn_s_wait_asynccnt(0);
#else
  asm volatile("s_wait_asynccnt 0" ::: "memory");
#endif
}
#endif

__device__ __forceinline__ float min5(float a, float b, float c, float d, float e) {
  return fminf(fminf(fminf(a, b), fminf(c, d)), e);
}

// One block: one 64x64 output tile of one image pair. Computes windowed
// channel-min for X and Y (35x35 window, +inf pad == -dark_channel), then
// accumulates sum(|dcx - dcy|) into partial[block].
__global__ __launch_bounds__(256) void dc_tile_kernel(const float* __restrict__ X,
                                                      const float* __restrict__ Y,
                                                      float* __restrict__ partial) {
  extern __shared__ float pool[];
  float* M  = pool;                    // EXT*EXT  channel-min tile (stride 98)
  float* A  = pool + EXT * EXT;        // EXT*EXT  raw stage / h-blocks / v-blocks
  float* Hm = pool + 2 * EXT * EXT;    // EXT*TILE horizontal window-min (stride 64)

  const int tid = threadIdx.x;
  const int ty = (int)(blockIdx.x >> 3) * TILE;
  const int tx = (int)(blockIdx.x & 7) * TILE;

  float dcx[16];
  float acc = 0.0f;

  for (int img = 0; img < 2; ++img) {
    const float* src = (img == 0 ? X : Y) + (size_t)blockIdx.y * 3 * HWsz;

    // ---------- Phase 1: channel-min into M (+inf outside image) ----------
#if HAVE_ASYNC
    for (int c = 0; c < 3; ++c) {
      const float* chan = src + (size_t)c * HWsz;
      for (int idx = tid; idx < EXT * EXT; idx += 256) {
        const int r = idx / EXT, q = idx - r * EXT;
        int cy = ty + r - PAD; cy = cy < 0 ? 0 : (cy > HH - 1 ? HH - 1 : cy);
        int cx = tx + q - PAD; cx = cx < 0 ? 0 : (cx > WW - 1 ? WW - 1 : cx);
        async_cp_b32(chan + cy * WW + cx, &A[idx]);  // DMA global -> LDS
      }
      wait_async0();
      __syncthreads();
      for (int idx = tid; idx < EXT * EXT; idx += 256) {
        const int r = idx / EXT, q = idx - r * EXT;
        const int gy = ty + r - PAD, gx = tx + q - PAD;
        const bool in = (gy >= 0) & (gy < HH) & (gx >= 0) & (gx < WW);
        const float v = in ? A[idx] : __builtin_inff();
        M[idx] = (c == 0) ? v : fminf(M[idx], v);
      }
      __syncthreads();
    }
#else
    for (int idx = tid; idx < EXT * EXT; idx += 256) {
      const int r = idx / EXT, q = idx - r * EXT;
      const int gy = ty + r - PAD, gx = tx + q - PAD;
      float v = __builtin_inff();
      if ((gy >= 0) & (gy < HH) & (gx >= 0) & (gx < WW)) {
        const float* p = src + gy * WW + gx;
        v = fminf(fminf(p[0], p[HWsz]), p[2 * HWsz]);
      }
      M[idx] = v;
    }
    __syncthreads();
#endif

    // ---------- Phase 2: horizontal 5-wide block mins (A <- hb) ----------
    for (int idx = tid; idx < EXT * HBW; idx += 256) {
      const int r = idx / HBW, q = idx - r * HBW;
      const float* m = &M[r * EXT + q];
      A[r * HBW + q] = min5(m[0], m[1], m[2], m[3], m[4]);
    }
    __syncthreads();

    // ---------- Phase 3: horizontal window-min (Hm, 35 = 7 blocks of 5) ----------
    for (int idx = tid; idx < EXT * TILE; idx += 256) {
      const int r = idx >> 6, j = idx & 63;
      const float* b = &A[r * HBW + j];
      float v = fminf(b[0], b[5]);
      v = fminf(v, b[10]); v = fminf(v, b[15]);
      v = fminf(v, b[20]); v = fminf(v, b[25]);
      v = fminf(v, b[30]);
      Hm[r * TILE + j] = v;
    }
    __syncthreads();

    // ---------- Phase 4: vertical 5-tall block mins (A <- vb, hb dead) ----------
    for (int idx = tid; idx < HBW * TILE; idx += 256) {
      const int i = idx >> 6, j = idx & 63;
      const float* h = &Hm[i * TILE + j];
      A[i * TILE + j] = min5(h[0], h[64], h[128], h[192], h[256]);
    }
    __syncthreads();

    // ---------- Phase 5: vertical window-min -> per-thread 16 outputs ----------
    {
      const int j = tid & 63;
      const int i0 = (tid >> 6) * 16;
#pragma unroll
      for (int ii = 0; ii < 16; ++ii) {
        const float* vb = &A[(i0 + ii) * TILE + j];
        float v = fminf(vb[0], vb[5 * TILE]);
        v = fminf(v, vb[10 * TILE]); v = fminf(v, vb[15 * TILE]);
        v = fminf(v, vb[20 * TILE]); v = fminf(v, vb[25 * TILE]);
        v = fminf(v, vb[30 * TILE]);
        if (img == 0) dcx[ii] = v;
        else          acc += fabsf(dcx[ii] - v);
      }
    }
    __syncthreads();  // LDS reused by next image / reduction
  }

  // ---------- Deterministic block reduction (wave32 shuffles + LDS) ----------
#pragma unroll
  for (int off = 16; off > 0; off >>= 1) acc += __shfl_down(acc, off, 32);
  if ((tid & 31) == 0) pool[tid >> 5] = acc;
  __syncthreads();
  if (tid == 0) {
    float s = 0.0f;
#pragma unroll
    for (int w = 0; w < 8; ++w) s += pool[w];
    partial[blockIdx.y * gridDim.x + blockIdx.x] = s;
  }
}

__global__ __launch_bounds__(256) void dc_finalize_kernel(const float* __restrict__ partial,
                                                          float* __restrict__ out) {
  __shared__ float sm[8];
  const int tid = threadIdx.x;
  float s = 0.0f;
  for (int i = tid; i < NPART; i += 256) s += partial[i];
#pragma unroll
  for (int off = 16; off > 0; off >>= 1) s += __shfl_down(s, off, 32);
  if ((tid & 31) == 0) sm[tid >> 5] = s;
  __syncthreads();
  if (tid == 0) {
    float t = 0.0f;
#pragma unroll
    for (int w = 0; w < 8; ++w) t += sm[w];
    out[0] = t * (1.0f / (16.0f * 512.0f * 512.0f));
  }
}

extern "C" void kernel_launch(void* const* d_in, const int* in_sizes, int n_in,
                              void* d_out, int out_size, void* d_ws, size_t ws_size,
                              hipStream_t stream) {
  (void)in_sizes; (void)n_in; (void)out_size; (void)ws_size;
  const float* X = (const float*)d_in[0];
  const float* Y = (const float*)d_in[1];
  float* partial = (float*)d_ws;          // NPART floats = 4 KB of scratch
  float* out = (float*)d_out;

  dim3 grid(64, 16);                      // 64 tiles per image pair, 16 pairs
  const size_t shmem = (size_t)(2 * EXT * EXT + EXT * TILE) * sizeof(float);  // ~102 KB
  dc_tile_kernel<<<grid, 256, shmem, stream>>>(X, Y, partial);
  dc_finalize_kernel<<<1, 256, 0, stream>>>(partial, out);
}